// GATNetwork_37641093382621
// MI455X (gfx1250) — compile-verified
//
#include <hip/hip_runtime.h>
#include <stdint.h>

// ---------------------------------------------------------------------------
// GAT network for MI455X (gfx1250, wave32).
//  - Dense projections via V_WMMA_F32_16X16X4_F32 (fp32 tensor path).
//  - Edge softmax/aggregation via L2-resident atomics (h/out fit in 192MB L2).
// ---------------------------------------------------------------------------

typedef float v2f __attribute__((ext_vector_type(2)));
typedef float v8f __attribute__((ext_vector_type(8)));

#define NEG_SLOPE 0.2f
#define N_NODES   50000
#define N_EDGES   800000
#define N_ETOT    850000   // edges + self loops
#define BATCH     1024

// ---------------- WMMA fp32 GEMM: C[M,N] = A[M,K] @ B[K,N] (+bias, relu) ----
// One wave per 16x16 output tile. All dims here are multiples of 16 (K mult 4).
__global__ void wmma_gemm_f32(const float* __restrict__ A,
                              const float* __restrict__ B,
                              const float* __restrict__ bias,
                              float* __restrict__ C,
                              int M, int K, int N, int relu)
{
    const int wave   = blockIdx.x * (blockDim.x >> 5) + (threadIdx.x >> 5);
    const int tilesN = N >> 4;
    const int tiles  = (M >> 4) * tilesN;
    if (wave >= tiles) return;                    // wave-uniform: EXEC stays all-1s
    const int tm = (wave / tilesN) << 4;
    const int tn = (wave % tilesN) << 4;
    const int lane = threadIdx.x & 31;
    const int r  = lane & 15;                     // M-row of A / N-col of B,C
    const int kh = lane >> 4;                     // K half-select (0: K0/1, 1: K2/3)

    v8f acc = {0.f,0.f,0.f,0.f,0.f,0.f,0.f,0.f};
    const float* Arow = A + (size_t)(tm + r) * K;
    for (int k0 = 0; k0 < K; k0 += 4) {
        const int ka = k0 + 2 * kh;
        v2f a, b;
        a.x = Arow[ka];
        a.y = Arow[ka + 1];
        b.x = B[(size_t)ka       * N + tn + r];
        b.y = B[(size_t)(ka + 1) * N + tn + r];
        // D = A(16x4) * B(4x16) + C  -> v_wmma_f32_16x16x4_f32
        acc = __builtin_amdgcn_wmma_f32_16x16x4_f32(
                  false, a, false, b, (short)0, acc, false, false);
    }
    const int col = tn + r;
    const float bv = bias ? bias[col] : 0.f;
    #pragma unroll
    for (int v = 0; v < 8; ++v) {
        float x = acc[v] + bv;
        if (relu) x = fmaxf(x, 0.f);
        C[(size_t)(tm + v + kh * 8) * N + col] = x;   // C/D layout: M = v + 8*kh
    }
}

// ---------------- per-node attention logits s,d: einsum('nhc,hc->nh') -------
__global__ void compute_sd(const float* __restrict__ h,
                           const float* __restrict__ a_src,
                           const float* __restrict__ a_dst,
                           float* __restrict__ s, float* __restrict__ d,
                           int Nn, int H, int C)
{
    const int tid = blockIdx.x * blockDim.x + threadIdx.x;
    if (tid >= Nn * H) return;
    const int n = tid / H, hh = tid - n * H;
    const float* hp = h + (size_t)n * H * C + hh * C;
    const float* as = a_src + hh * C;
    const float* ad = a_dst + hh * C;
    float ss = 0.f, dd = 0.f;
    for (int c = 0; c < C; ++c) { const float v = hp[c]; ss += v * as[c]; dd += v * ad[c]; }
    s[tid] = ss; d[tid] = dd;
}

// monotone float<->uint encoding so segment-max can use u32 atomicMax
__device__ __forceinline__ unsigned fenc(float f) {
    unsigned u = __float_as_uint(f);
    return (u & 0x80000000u) ? ~u : (u | 0x80000000u);
}
__device__ __forceinline__ float fdec(unsigned u) {
    return (u & 0x80000000u) ? __uint_as_float(u & 0x7FFFFFFFu) : __uint_as_float(~u);
}

__device__ __forceinline__ void edge_ends(const int* __restrict__ ei, int eid,
                                          int& sN, int& dN)
{
    if (eid < N_EDGES) { sN = ei[eid]; dN = ei[N_EDGES + eid]; }
    else               { sN = dN = eid - N_EDGES; }         // appended self loop
}

// pass 1: m[dst,h] = max over incoming edges of leaky(s[src]+d[dst])
__global__ void edge_max(const int* __restrict__ ei,
                         const float* __restrict__ s, const float* __restrict__ d,
                         unsigned* __restrict__ menc, int H)
{
    const int tid = blockIdx.x * blockDim.x + threadIdx.x;
    if (tid >= N_ETOT * H) return;
    const int eid = tid / H, hh = tid - eid * H;
    int sN, dN; edge_ends(ei, eid, sN, dN);
    float e = s[sN * H + hh] + d[dN * H + hh];
    e = (e > 0.f) ? e : NEG_SLOPE * e;
    atomicMax(&menc[dN * H + hh], fenc(e));
}

__global__ void decode_max(unsigned* __restrict__ m, int n)
{
    const int t = blockIdx.x * blockDim.x + threadIdx.x;
    if (t < n) m[t] = __float_as_uint(fdec(m[t]));
}

// pass 2: denom[dst,h] = sum exp(e - m[dst,h])
__global__ void edge_denom(const int* __restrict__ ei,
                           const float* __restrict__ s, const float* __restrict__ d,
                           const float* __restrict__ m, float* __restrict__ denom, int H)
{
    const int tid = blockIdx.x * blockDim.x + threadIdx.x;
    if (tid >= N_ETOT * H) return;
    const int eid = tid / H, hh = tid - eid * H;
    int sN, dN; edge_ends(ei, eid, sN, dN);
    float e = s[sN * H + hh] + d[dN * H + hh];
    e = (e > 0.f) ? e : NEG_SLOPE * e;
    atomicAdd(&denom[dN * H + hh], __expf(e - m[dN * H + hh]));
}

// pass 3: out[dst,:] += alpha * h[src,:]   (64 threads per edge; F=H*C)
__global__ void edge_agg(const int* __restrict__ ei, const float* __restrict__ h,
                         const float* __restrict__ s, const float* __restrict__ d,
                         const float* __restrict__ m, const float* __restrict__ denom,
                         float* __restrict__ out, int H, int C)
{
    const int eid = blockIdx.x * (blockDim.x >> 6) + (threadIdx.x >> 6);
    if (eid >= N_ETOT) return;
    const int f0 = threadIdx.x & 63;
    const int F  = H * C;
    int sN, dN; edge_ends(ei, eid, sN, dN);
    for (int f = f0; f < F; f += 64) {
        const int hh = f / C;
        float e = s[sN * H + hh] + d[dN * H + hh];
        e = (e > 0.f) ? e : NEG_SLOPE * e;
        const float alpha = __expf(e - m[dN * H + hh]) / denom[dN * H + hh];
        atomicAdd(&out[(size_t)dN * F + f], h[(size_t)sN * F + f] * alpha);
    }
}

__global__ void bias_act(float* __restrict__ out, const float* __restrict__ b,
                         int total, int F, int relu)
{
    const int t = blockIdx.x * blockDim.x + threadIdx.x;
    if (t >= total) return;
    float x = out[t] + b[t % F];
    if (relu) x = fmaxf(x, 0.f);
    out[t] = x;
}

__global__ void softmax16(const float* __restrict__ in, float* __restrict__ out, int rows)
{
    const int r = blockIdx.x * blockDim.x + threadIdx.x;
    if (r >= rows) return;
    const float* p = in + (size_t)r * 16;
    float mx = -3.0e38f;
    #pragma unroll
    for (int i = 0; i < 16; ++i) mx = fmaxf(mx, p[i]);
    float e[16], sum = 0.f;
    #pragma unroll
    for (int i = 0; i < 16; ++i) { e[i] = __expf(p[i] - mx); sum += e[i]; }
    const float inv = 1.f / sum;
    #pragma unroll
    for (int i = 0; i < 16; ++i) out[(size_t)r * 16 + i] = e[i] * inv;
}

// ---------------------------------------------------------------------------
static inline int cdiv(int a, int b) { return (a + b - 1) / b; }

extern "C" void kernel_launch(void* const* d_in, const int* in_sizes, int n_in,
                              void* d_out, int out_size, void* d_ws, size_t ws_size,
                              hipStream_t stream)
{
    const float* x    = (const float*)d_in[0];
    const int*   ei   = (const int*)  d_in[1];   // edge_index (2,800000), int32
    // d_in[2] = batch_size (1024)
    const float* W1   = (const float*)d_in[3];
    const float* as1  = (const float*)d_in[4];
    const float* ad1  = (const float*)d_in[5];
    const float* b1   = (const float*)d_in[6];
    const float* W2   = (const float*)d_in[7];
    const float* as2  = (const float*)d_in[8];
    const float* ad2  = (const float*)d_in[9];
    const float* b2   = (const float*)d_in[10];
    const float* W3   = (const float*)d_in[11];
    const float* as3  = (const float*)d_in[12];
    const float* ad3  = (const float*)d_in[13];
    const float* b3   = (const float*)d_in[14];
    const float* Wm1  = (const float*)d_in[15];
    const float* bm1  = (const float*)d_in[16];
    const float* Wm2  = (const float*)d_in[17];
    const float* bm2  = (const float*)d_in[18];

    // workspace layout
    const size_t NB = (size_t)N_NODES * 256 * sizeof(float);   // 51.2 MB
    char* w = (char*)d_ws;
    float*    bufA  = (float*)w;  w += NB;                     // h = in @ W
    float*    bufB  = (float*)w;  w += NB;                     // aggregated out / next in
    float*    sbuf  = (float*)w;  w += (size_t)N_NODES * 4 * sizeof(float);
    float*    dbuf  = (float*)w;  w += (size_t)N_NODES * 4 * sizeof(float);
    unsigned* menc  = (unsigned*)w; w += (size_t)N_NODES * 4 * sizeof(unsigned);
    float*    denom = (float*)w;  w += (size_t)N_NODES * 4 * sizeof(float);
    float*    mid   = (float*)w;  w += (size_t)BATCH * 64 * sizeof(float);
    float*    out2  = (float*)w;  w += (size_t)BATCH * 16 * sizeof(float);

    auto gat_layer = [&](const float* in, int Kin, int H, int C,
                         const float* W, const float* as_, const float* ad_,
                         const float* bias, int relu) {
        const int F = H * C;
        // h = in @ W  (WMMA fp32)
        {
            const int tiles = (N_NODES / 16) * (F / 16);
            wmma_gemm_f32<<<cdiv(tiles, 8), 256, 0, stream>>>(in, W, nullptr, bufA,
                                                              N_NODES, Kin, F, 0);
        }
        compute_sd<<<cdiv(N_NODES * H, 256), 256, 0, stream>>>(bufA, as_, ad_, sbuf, dbuf,
                                                               N_NODES, H, C);
        hipMemsetAsync(menc,  0, (size_t)N_NODES * H * 4, stream);  // 0 < fenc(-inf) = 0x007FFFFF
        hipMemsetAsync(denom, 0, (size_t)N_NODES * H * 4, stream);
        edge_max  <<<cdiv(N_ETOT * H, 256), 256, 0, stream>>>(ei, sbuf, dbuf, menc, H);
        decode_max<<<cdiv(N_NODES * H, 256), 256, 0, stream>>>(menc, N_NODES * H);
        edge_denom<<<cdiv(N_ETOT * H, 256), 256, 0, stream>>>(ei, sbuf, dbuf,
                                                              (const float*)menc, denom, H);
        hipMemsetAsync(bufB, 0, (size_t)N_NODES * F * sizeof(float), stream);
        edge_agg  <<<cdiv(N_ETOT, 4), 256, 0, stream>>>(ei, bufA, sbuf, dbuf,
                                                        (const float*)menc, denom, bufB, H, C);
        bias_act  <<<cdiv(N_NODES * F, 256), 256, 0, stream>>>(bufB, bias, N_NODES * F, F, relu);
    };

    gat_layer(x,    128, 4, 64, W1, as1, ad1, b1, 1);
    gat_layer(bufB, 256, 4, 64, W2, as2, ad2, b2, 1);
    gat_layer(bufB, 256, 1, 64, W3, as3, ad3, b3, 0);

    // MLP head on the first 1024 rows only (row-wise ops commute with the slice)
    wmma_gemm_f32<<<cdiv((BATCH / 16) * (64 / 16), 8), 256, 0, stream>>>(
        bufB, Wm1, bm1, mid, BATCH, 64, 64, 1);
    wmma_gemm_f32<<<cdiv((BATCH / 16) * (16 / 16), 8), 256, 0, stream>>>(
        mid, Wm2, bm2, out2, BATCH, 64, 16, 0);
    softmax16<<<cdiv(BATCH, 256), 256, 0, stream>>>(out2, (float*)d_out, BATCH);
}